// StructureExtractor_16904991277430
// MI455X (gfx1250) — compile-verified
//
#include <hip/hip_runtime.h>

#define D 128
#define EPSV 1e-5f

typedef __attribute__((ext_vector_type(2))) float v2f;
typedef __attribute__((ext_vector_type(8))) float v8f;

// ---------------- small elementwise kernels ----------------

__global__ void zero_f32(float* __restrict__ p, int n) {
  int i = blockIdx.x * blockDim.x + threadIdx.x;
  if (i < n) p[i] = 0.0f;
}

__global__ void degree_kernel(const long long* __restrict__ dst, int E,
                              float* __restrict__ deg) {
  int e = blockIdx.x * blockDim.x + threadIdx.x;
  if (e < E) atomicAdd(&deg[(int)dst[e]], 1.0f);
}

__global__ void dinv_kernel(float* __restrict__ deg, int n) {
  int i = blockIdx.x * blockDim.x + threadIdx.x;
  if (i < n) deg[i] = rsqrtf(deg[i] + 1.0f);  // +1 self loop
}

// agg[n,:] = h[n,:] * dinv[n]^2   (self-loop message; also zero-bases agg)
__global__ void selfloop_init(const float* __restrict__ h,
                              const float* __restrict__ dinv,
                              float* __restrict__ agg, int n) {
  int i = blockIdx.x * blockDim.x + threadIdx.x;
  if (i < n * D) {
    float dv = dinv[i >> 7];
    agg[i] = h[i] * dv * dv;
  }
}

// one wave per edge: lane covers 4 floats -> 32 lanes * 4 = 128 = D
__global__ __launch_bounds__(256) void edge_scatter(
    const long long* __restrict__ src, const long long* __restrict__ dst,
    const float* __restrict__ dinv, const float* __restrict__ h,
    float* __restrict__ agg, int E) {
  int gid = blockIdx.x * blockDim.x + threadIdx.x;
  int e = gid >> 5;
  int lane = gid & 31;
  if (e >= E) return;
  int s = (int)src[e];
  int d = (int)dst[e];
  float norm = dinv[s] * dinv[d];
  const float4 hv = *(const float4*)&h[s * D + lane * 4];
  float* ap = &agg[d * D + lane * 4];
  atomicAdd(ap + 0, hv.x * norm);
  atomicAdd(ap + 1, hv.y * norm);
  atomicAdd(ap + 2, hv.z * norm);
  atomicAdd(ap + 3, hv.w * norm);
}

__global__ void bias_relu(const float* __restrict__ agg,
                          const float* __restrict__ b,
                          float* __restrict__ out, int n) {
  int i = blockIdx.x * blockDim.x + threadIdx.x;
  if (i < n * D) {
    float v = agg[i] + b[i & (D - 1)];
    out[i] = v > 0.0f ? v : 0.0f;
  }
}

// ---------------- WMMA GEMM: C[N,128] = A[N,128] @ W[128,128] ----------------
// block = 256 threads = 8 waves; block tile 16 rows x 128 cols; wave tile 16x16.
__global__ __launch_bounds__(256) void gemm128_wmma(
    const float* __restrict__ A, const float* __restrict__ W,
    float* __restrict__ C, int nrows) {
  __shared__ float Xs[16][33];     // 16 rows x 32 K (padded)
  __shared__ float Ws[32][128];    // 32 K x 128 cols
  const int tid = threadIdx.x;
  const int wave = tid >> 5;
  const int lane = tid & 31;
  const int half = lane >> 4;
  const int l = lane & 15;
  const int row0 = blockIdx.x * 16;
  const int col0 = wave * 16;
  v8f acc = {};
  for (int k0 = 0; k0 < 128; k0 += 32) {
    for (int i = tid; i < 16 * 32; i += 256) {
      int r = i >> 5, c = i & 31;
      int gr = row0 + r;
      Xs[r][c] = (gr < nrows) ? A[gr * D + k0 + c] : 0.0f;
    }
    for (int i = tid; i < 32 * 32; i += 256) {
      int r = i >> 5, c4 = (i & 31) * 4;
      *(float4*)&Ws[r][c4] = *(const float4*)&W[(k0 + r) * D + c4];
    }
    __syncthreads();
#pragma unroll
    for (int kk = 0; kk < 32; kk += 4) {
      v2f a, b;
      a.x = Xs[l][kk + 2 * half + 0];
      a.y = Xs[l][kk + 2 * half + 1];
      b.x = Ws[kk + 2 * half + 0][col0 + l];
      b.y = Ws[kk + 2 * half + 1][col0 + l];
      acc = __builtin_amdgcn_wmma_f32_16x16x4_f32(false, a, false, b,
                                                  (short)0, acc, false, false);
    }
    __syncthreads();
  }
#pragma unroll
  for (int r = 0; r < 8; ++r) {
    int row = row0 + r + 8 * half;
    if (row < nrows) C[row * D + col0 + l] = acc[r];
  }
}

// ---------------- BatchNorm statistics ----------------
// grid = (128, 3); blockIdx.y selects source (x | h1 | h2). Coalesced reads.
__global__ __launch_bounds__(256) void bn_stats(
    const float* __restrict__ x, const float* __restrict__ h1,
    const float* __restrict__ h2, float* __restrict__ sum,
    float* __restrict__ sumsq, int n) {
  const float* S = (blockIdx.y == 0) ? x : (blockIdx.y == 1) ? h1 : h2;
  int col = threadIdx.x & (D - 1);
  int rh = threadIdx.x >> 7;  // 0 or 1
  float s = 0.0f, s2 = 0.0f;
  for (int r = blockIdx.x * 2 + rh; r < n; r += gridDim.x * 2) {
    float v = S[r * D + col];
    s += v;
    s2 += v * v;
  }
  __shared__ float ls[D], ls2[D];
  if (rh) { ls[col] = s; ls2[col] = s2; }
  __syncthreads();
  if (!rh) {
    s += ls[col];
    s2 += ls2[col];
    atomicAdd(&sum[blockIdx.y * D + col], s);
    atomicAdd(&sumsq[blockIdx.y * D + col], s2);
  }
}

// Fold BN into output linear: Wf[c,j] = s_c*Wo[c,j];
// biasp[j] = bo[j] + sum_c (beta[c]-mu[c]*s_c)*Wo[c,j]
__global__ void bn_finalize(const float* __restrict__ sum,
                            const float* __restrict__ sumsq,
                            const float* __restrict__ gamma,
                            const float* __restrict__ beta,
                            const float* __restrict__ Wo,
                            const float* __restrict__ bo,
                            float* __restrict__ Wf,
                            float* __restrict__ biasp, int n) {
  int j = threadIdx.x;  // 128 threads, single block
  float invn = 1.0f / (float)n;
  float bp = 0.0f;
  for (int c = 0; c < 3 * D; ++c) {
    float mu = sum[c] * invn;
    float var = sumsq[c] * invn - mu * mu;
    float sc = gamma[c] * rsqrtf(var + EPSV);
    float w = Wo[c * D + j];
    Wf[c * D + j] = sc * w;
    bp += (beta[c] - mu * sc) * w;
  }
  biasp[j] = bo[j] + bp;
}

// ---------------- final GEMM: out = [x|h1|h2] @ Wf + biasp ----------------
__global__ __launch_bounds__(256) void final_gemm_wmma(
    const float* __restrict__ x, const float* __restrict__ h1,
    const float* __restrict__ h2, const float* __restrict__ Wf,
    const float* __restrict__ biasp, float* __restrict__ out, int nrows) {
  __shared__ float Xs[16][33];
  __shared__ float Ws[32][128];
  const int tid = threadIdx.x;
  const int wave = tid >> 5;
  const int lane = tid & 31;
  const int half = lane >> 4;
  const int l = lane & 15;
  const int row0 = blockIdx.x * 16;
  const int col0 = wave * 16;
  v8f acc = {};
  for (int k0 = 0; k0 < 3 * D; k0 += 32) {
    const float* A = (k0 < D) ? x : (k0 < 2 * D) ? h1 : h2;
    const int kb = k0 & (D - 1);
    for (int i = tid; i < 16 * 32; i += 256) {
      int r = i >> 5, c = i & 31;
      int gr = row0 + r;
      Xs[r][c] = (gr < nrows) ? A[gr * D + kb + c] : 0.0f;
    }
    for (int i = tid; i < 32 * 32; i += 256) {
      int r = i >> 5, c4 = (i & 31) * 4;
      *(float4*)&Ws[r][c4] = *(const float4*)&Wf[(k0 + r) * D + c4];
    }
    __syncthreads();
#pragma unroll
    for (int kk = 0; kk < 32; kk += 4) {
      v2f a, b;
      a.x = Xs[l][kk + 2 * half + 0];
      a.y = Xs[l][kk + 2 * half + 1];
      b.x = Ws[kk + 2 * half + 0][col0 + l];
      b.y = Ws[kk + 2 * half + 1][col0 + l];
      acc = __builtin_amdgcn_wmma_f32_16x16x4_f32(false, a, false, b,
                                                  (short)0, acc, false, false);
    }
    __syncthreads();
  }
  float bj = biasp[col0 + l];
#pragma unroll
  for (int r = 0; r < 8; ++r) {
    int row = row0 + r + 8 * half;
    if (row < nrows) out[row * D + col0 + l] = acc[r] + bj;
  }
}

// ---------------- launch ----------------

extern "C" void kernel_launch(void* const* d_in, const int* in_sizes, int n_in,
                              void* d_out, int out_size, void* d_ws,
                              size_t ws_size, hipStream_t stream) {
  const float* x = (const float*)d_in[0];
  const long long* ei = (const long long*)d_in[1];
  const float* W1 = (const float*)d_in[2];
  const float* b1 = (const float*)d_in[3];
  const float* W2 = (const float*)d_in[4];
  const float* b2 = (const float*)d_in[5];
  const float* gamma = (const float*)d_in[6];
  const float* beta = (const float*)d_in[7];
  const float* Wo = (const float*)d_in[8];
  const float* bo = (const float*)d_in[9];
  float* out = (float*)d_out;

  const int N = in_sizes[0] / D;
  const int E = in_sizes[1] / 2;
  const long long* src = ei;
  const long long* dst = ei + E;

  // workspace carve-out (floats), 256B-aligned chunks
  float* ws = (float*)d_ws;
  size_t off = 0;
  auto carve = [&](size_t nfloats) {
    float* p = ws + off;
    off += (nfloats + 63) & ~(size_t)63;
    return p;
  };
  float* deg   = carve((size_t)N);            // deg -> dinv in place
  float* sum   = carve(3 * D);                // BN sums
  float* sumsq = carve(3 * D);
  float* Wf    = carve((size_t)3 * D * D);    // folded output weight
  float* biasp = carve(D);                    // folded output bias
  float* hbuf  = carve((size_t)N * D);        // x@W (reused both layers)
  float* agg   = carve((size_t)N * D);        // aggregation (reused)
  float* h1    = carve((size_t)N * D);
  float* h2    = carve((size_t)N * D);
  (void)ws_size; (void)n_in; (void)out_size;

  const int T = 256;
  const int ndBlocks = (N * D + T - 1) / T;
  const int gemmBlocks = (N + 15) / 16;

  // degrees -> dinv
  zero_f32<<<(N + T - 1) / T, T, 0, stream>>>(deg, N);
  zero_f32<<<(6 * D + T - 1) / T, T, 0, stream>>>(sum, 6 * D);  // sum+sumsq contiguous
  degree_kernel<<<(E + T - 1) / T, T, 0, stream>>>(dst, E, deg);
  dinv_kernel<<<(N + T - 1) / T, T, 0, stream>>>(deg, N);

  // layer 1
  gemm128_wmma<<<gemmBlocks, T, 0, stream>>>(x, W1, hbuf, N);
  selfloop_init<<<ndBlocks, T, 0, stream>>>(hbuf, deg, agg, N);
  edge_scatter<<<((size_t)E * 32 + T - 1) / T, T, 0, stream>>>(src, dst, deg, hbuf, agg, E);
  bias_relu<<<ndBlocks, T, 0, stream>>>(agg, b1, h1, N);

  // layer 2
  gemm128_wmma<<<gemmBlocks, T, 0, stream>>>(h1, W2, hbuf, N);
  selfloop_init<<<ndBlocks, T, 0, stream>>>(hbuf, deg, agg, N);
  edge_scatter<<<((size_t)E * 32 + T - 1) / T, T, 0, stream>>>(src, dst, deg, hbuf, agg, E);
  bias_relu<<<ndBlocks, T, 0, stream>>>(agg, b2, h2, N);

  // batchnorm stats + fold into output linear
  bn_stats<<<dim3(128, 3), T, 0, stream>>>(x, h1, h2, sum, sumsq, N);
  bn_finalize<<<1, D, 0, stream>>>(sum, sumsq, gamma, beta, Wo, bo, Wf, biasp, N);

  // out = [x|h1|h2] @ Wf + biasp
  final_gemm_wmma<<<gemmBlocks, T, 0, stream>>>(x, h1, h2, Wf, biasp, out, N);
}